// GNN_77764677861850
// MI455X (gfx1250) — compile-verified
//
#include <hip/hip_runtime.h>
#include <hip/hip_bf16.h>

#define NN 50000
#define NE 800000
#define FEAT 128
#define NG 128
#define OUTC 10
#define BN_EPS 1e-5f

typedef float v2f __attribute__((ext_vector_type(2)));
typedef float v8f __attribute__((ext_vector_type(8)));

// ---------------- degree kernels ----------------
__global__ void edge_degree_kernel(const int* __restrict__ src, const int* __restrict__ dst,
                                   float* __restrict__ deg_out, float* __restrict__ deg_in) {
    int e = blockIdx.x * blockDim.x + threadIdx.x;
    if (e >= NE) return;
    atomicAdd(&deg_out[src[e]], 1.0f);
    atomicAdd(&deg_in[dst[e]], 1.0f);
}

__global__ void invsqrt_kernel(const float* __restrict__ deg_out, const float* __restrict__ deg_in,
                               float* __restrict__ inv_out, float* __restrict__ inv_in) {
    int i = blockIdx.x * blockDim.x + threadIdx.x;
    if (i >= NN) return;
    inv_out[i] = rsqrtf(fmaxf(deg_out[i], 1.0f));
    inv_in[i]  = rsqrtf(fmaxf(deg_in[i], 1.0f));
}

// ---------------- WMMA GEMM, cols fixed at 128 (compile-time strides) ----------------
// out[rows x 128] = (scaleA ? diag(scaleA) : I) * A[rows x 128] @ W[128 x 128] (+bias)(+relu)
// block = 256 threads = 8 waves; block handles 16 rows; wave w handles cols [16w,16w+16).
// V_WMMA_F32_16X16X4_F32, fully unrolled 32-step K loop, no divergence anywhere.
__global__ void gemm128_wmma_kernel(const float* __restrict__ A,
                                    const float* __restrict__ W,
                                    float* __restrict__ out,
                                    const float* __restrict__ scaleA,
                                    const float* __restrict__ bias,
                                    int relu) {
    const int lane = threadIdx.x & 31;
    const int wave = threadIdx.x >> 5;
    const int r0 = blockIdx.x * 16;
    const int c0 = wave * 16;

    const int m  = lane & 15;            // A row within tile (ISA 16x4 f32 layout)
    const int kb = (lane >> 4) << 1;     // lanes 0-15 -> K{0,1}, lanes 16-31 -> K{2,3}
    const int n  = lane & 15;            // B/D column within tile

    const float sA = scaleA ? scaleA[r0 + m] : 1.0f;
    const float* arow = A + (size_t)(r0 + m) * FEAT + kb;      // 8B-aligned (kb even)
    const float* wcol = W + (size_t)kb * FEAT + c0 + n;        // stride FEAT between K rows

    v8f acc = {0.f, 0.f, 0.f, 0.f, 0.f, 0.f, 0.f, 0.f};
#pragma unroll
    for (int k0 = 0; k0 < FEAT; k0 += 4) {
        v2f a = *(const v2f*)(arow + k0);
        a.x *= sA;
        a.y *= sA;
        v2f b;
        b.x = wcol[(size_t)k0 * FEAT];
        b.y = wcol[(size_t)(k0 + 1) * FEAT];
        acc = __builtin_amdgcn_wmma_f32_16x16x4_f32(
            /*neg_a=*/false, a, /*neg_b=*/false, b,
            /*c_mod=*/(short)0, acc, /*reuse_a=*/false, /*reuse_b=*/false);
    }

    const float bv = bias ? bias[c0 + n] : 0.0f;
    const int rhalf = (lane >> 4) * 8;   // D: VGPR r -> lanes 0-15 M=r, lanes 16-31 M=r+8
#pragma unroll
    for (int r = 0; r < 8; ++r) {
        float v = acc[r] + bv;
        if (relu) v = fmaxf(v, 0.0f);
        out[(size_t)(r0 + rhalf + r) * FEAT + c0 + n] = v;
    }
}

// ---------------- edge scatter-add: agg[dst[e]] += h[src[e]] ----------------
// 32 lanes per edge, 4 floats per lane (float4 gather, 4 float L2 atomics).
__global__ void scatter_add_kernel(const float* __restrict__ h,
                                   const int* __restrict__ src, const int* __restrict__ dst,
                                   float* __restrict__ agg) {
    unsigned t = blockIdx.x * blockDim.x + threadIdx.x;
    unsigned e = t >> 5;
    if (e >= NE) return;
    int c = (t & 31) * 4;
    const float4 v = *(const float4*)(h + (size_t)src[e] * FEAT + c);
    float* o = agg + (size_t)dst[e] * FEAT + c;
    atomicAdd(o + 0, v.x);
    atomicAdd(o + 1, v.y);
    atomicAdd(o + 2, v.z);
    atomicAdd(o + 3, v.w);
}

// ---------------- x = relu(agg * inv_sqrt_in + b) ----------------
__global__ void finalize_kernel(const float* __restrict__ agg, const float* __restrict__ inv_in,
                                const float* __restrict__ bias, float* __restrict__ out) {
    unsigned t = blockIdx.x * blockDim.x + threadIdx.x;
    if (t >= (unsigned)NN * FEAT) return;
    unsigned nIdx = t >> 7;      // /128
    unsigned c = t & 127;
    float v = agg[t] * inv_in[nIdx] + bias[c];
    out[t] = fmaxf(v, 0.0f);
}

// ---------------- sum-pool into d_out embedding region ----------------
__global__ void pool_kernel(const float* __restrict__ x, const int* __restrict__ graph_ids,
                            float* __restrict__ emb) {
    unsigned t = blockIdx.x * blockDim.x + threadIdx.x;
    unsigned nIdx = t >> 5;
    if (nIdx >= NN) return;
    int c = (t & 31) * 4;
    int g = graph_ids[nIdx];
    const float4 v = *(const float4*)(x + (size_t)nIdx * FEAT + c);
    float* o = emb + (size_t)g * FEAT + c;
    atomicAdd(o + 0, v.x);
    atomicAdd(o + 1, v.y);
    atomicAdd(o + 2, v.z);
    atomicAdd(o + 3, v.w);
}

// ---------------- BatchNorm1d (batch stats, biased var), per-column thread ----------------
__global__ void bn_kernel(const float* __restrict__ emb, const float* __restrict__ gamma,
                          const float* __restrict__ beta, float* __restrict__ ybn) {
    int c = threadIdx.x;
    if (c >= FEAT) return;
    float mu = 0.f;
    for (int g = 0; g < NG; ++g) mu += emb[g * FEAT + c];
    mu *= (1.0f / NG);
    float var = 0.f;
    for (int g = 0; g < NG; ++g) {
        float d = emb[g * FEAT + c] - mu;
        var += d * d;
    }
    var *= (1.0f / NG);
    float s = gamma[c] * rsqrtf(var + BN_EPS);
    float b = beta[c];
    for (int g = 0; g < NG; ++g)
        ybn[g * FEAT + c] = (emb[g * FEAT + c] - mu) * s + b;
}

// ---------------- fused fc2 (128x10) + log_softmax, one thread per graph ----------------
__global__ void fc2_logsoftmax_kernel(const float* __restrict__ h1,
                                      const float* __restrict__ fc2_w,  // [128 x 10]
                                      const float* __restrict__ fc2_b,
                                      float* __restrict__ out) {
    int g = threadIdx.x;
    if (g >= NG) return;
    float logit[OUTC];
#pragma unroll
    for (int j = 0; j < OUTC; ++j) logit[j] = fc2_b[j];
    const float* hrow = h1 + (size_t)g * FEAT;
    for (int k = 0; k < FEAT; ++k) {
        float hv = hrow[k];
        const float* wrow = fc2_w + (size_t)k * OUTC;
#pragma unroll
        for (int j = 0; j < OUTC; ++j) logit[j] = fmaf(hv, wrow[j], logit[j]);
    }
    float mx = -1e30f;
#pragma unroll
    for (int j = 0; j < OUTC; ++j) mx = fmaxf(mx, logit[j]);
    float s = 0.f;
#pragma unroll
    for (int j = 0; j < OUTC; ++j) s += __expf(logit[j] - mx);
    float lse = mx + __logf(s);
#pragma unroll
    for (int j = 0; j < OUTC; ++j) out[g * OUTC + j] = logit[j] - lse;
}

extern "C" void kernel_launch(void* const* d_in, const int* in_sizes, int n_in,
                              void* d_out, int out_size, void* d_ws, size_t ws_size,
                              hipStream_t stream) {
    const float* n_feat    = (const float*)d_in[0];
    const int*   src       = (const int*)  d_in[1];
    const int*   dst       = (const int*)  d_in[2];
    const int*   graph_ids = (const int*)  d_in[3];
    const float* W1        = (const float*)d_in[4];
    const float* b1        = (const float*)d_in[5];
    const float* W2        = (const float*)d_in[6];
    const float* b2        = (const float*)d_in[7];
    const float* gamma     = (const float*)d_in[8];
    const float* beta      = (const float*)d_in[9];
    const float* fc1_w     = (const float*)d_in[10];
    const float* fc1_b     = (const float*)d_in[11];
    const float* fc2_w     = (const float*)d_in[12];
    const float* fc2_b     = (const float*)d_in[13];

    float* out = (float*)d_out;               // [0,16384): embedding, [16384,17664): log_softmax

    float* ws      = (float*)d_ws;
    float* deg_out = ws;                      // NN
    float* deg_in  = deg_out + NN;            // NN
    float* inv_out = deg_in  + NN;            // NN
    float* inv_in  = inv_out + NN;            // NN
    float* bufA    = inv_in  + NN;            // NN*FEAT
    float* bufB    = bufA + (size_t)NN * FEAT;// NN*FEAT
    float* ybn     = bufB + (size_t)NN * FEAT;// NG*FEAT
    float* h1      = ybn  + NG * FEAT;        // NG*FEAT

    // zero: degrees, embedding region of d_out
    hipMemsetAsync(deg_out, 0, (size_t)2 * NN * sizeof(float), stream);
    hipMemsetAsync(out, 0, (size_t)NG * FEAT * sizeof(float), stream);

    // degrees -> inv sqrt
    edge_degree_kernel<<<(NE + 255) / 256, 256, 0, stream>>>(src, dst, deg_out, deg_in);
    invsqrt_kernel<<<(NN + 255) / 256, 256, 0, stream>>>(deg_out, deg_in, inv_out, inv_in);

    const int nodeTiles = NN / 16;            // 3125 (exact)
    const int scatterBlocks = (NE * 32) / 256;// 100000 (exact)
    const int elemBlocks = (NN * FEAT) / 256; // 25000 (exact)

    // ---- layer 1: h = (x * inv_out) @ W1 ; agg = scatter ; x1 = relu(agg*inv_in + b1)
    gemm128_wmma_kernel<<<nodeTiles, 256, 0, stream>>>(n_feat, W1, bufA, inv_out, nullptr, 0);
    hipMemsetAsync(bufB, 0, (size_t)NN * FEAT * sizeof(float), stream);
    scatter_add_kernel<<<scatterBlocks, 256, 0, stream>>>(bufA, src, dst, bufB);
    finalize_kernel<<<elemBlocks, 256, 0, stream>>>(bufB, inv_in, b1, bufA);   // x1 -> bufA

    // ---- layer 2
    gemm128_wmma_kernel<<<nodeTiles, 256, 0, stream>>>(bufA, W2, bufB, inv_out, nullptr, 0);
    hipMemsetAsync(bufA, 0, (size_t)NN * FEAT * sizeof(float), stream);
    scatter_add_kernel<<<scatterBlocks, 256, 0, stream>>>(bufB, src, dst, bufA);
    finalize_kernel<<<elemBlocks, 256, 0, stream>>>(bufA, inv_in, b2, bufB);   // x2 -> bufB

    // ---- sum pooling into d_out embedding region
    pool_kernel<<<(NN * 32) / 256, 256, 0, stream>>>(bufB, graph_ids, out);

    // ---- head: BN -> fc1(+relu, WMMA) -> fused fc2+log_softmax
    bn_kernel<<<1, FEAT, 0, stream>>>(out, gamma, beta, ybn);
    gemm128_wmma_kernel<<<NG / 16, 256, 0, stream>>>(ybn, fc1_w, h1, nullptr, fc1_b, 1);
    fc2_logsoftmax_kernel<<<1, NG, 0, stream>>>(h1, fc2_w, fc2_b, out + (size_t)NG * FEAT);
}